// Almtx_59253368815831
// MI455X (gfx1250) — compile-verified
//
#include <hip/hip_runtime.h>
#include <hip/hip_bf16.h>

// Problem constants from the reference: b=8,k=4 -> R=32 rows, l=8192, d=128, 2d=256, S=64.
#define RROWS 32
#define LLEN  8192
#define DIN   128
#define DHID  256
#define NSEG  64

typedef __attribute__((ext_vector_type(2))) float v2f;
typedef __attribute__((ext_vector_type(8))) float v8f;

// ---------------------------------------------------------------------------
// K1: score -> g = |relu(X@W1+b1)@W2| using V_WMMA_F32_16X16X4_F32.
// One wave owns a 16-row M-tile. A-tile (16x128 f32) preloaded into VGPRs with
// the ISA 32-bit A layout: lane<16 holds K = 4*kk + {0,1}; lane>=16 holds
// K = 4*kk + {2,3} for row M = lane&15. B mirrors the same K split per lane.
// ---------------------------------------------------------------------------
__global__ __launch_bounds__(128) void mlp_score_kernel(
    const float* __restrict__ X,    // (P, 128) row-major
    const float* __restrict__ W1,   // (128, 256) row-major
    const float* __restrict__ B1,   // (256)
    const float* __restrict__ W2,   // (256)
    float* __restrict__ g_out,      // (P)
    int P)
{
    const int lane  = threadIdx.x & 31;
    const int wave  = threadIdx.x >> 5;
    const int tile  = blockIdx.x * 4 + wave;
    const int m0    = tile * 16;
    if (m0 >= P) return;                  // wave-uniform: EXEC stays all-ones inside

    const int mrow  = lane & 15;
    const int khalf = (lane >> 4) * 2;    // 0 (lanes 0-15) or 2 (lanes 16-31)
    const int ncol  = lane & 15;

    // Preload this wave's 16x128 A tile (64 VGPRs).
    v2f a[32];
    const float* xrow = X + (size_t)(m0 + mrow) * DIN + khalf;
#pragma unroll
    for (int kk = 0; kk < 32; ++kk) {
        a[kk].x = xrow[kk * 4 + 0];
        a[kk].y = xrow[kk * 4 + 1];
    }

    float part[8];
#pragma unroll
    for (int v = 0; v < 8; ++v) part[v] = 0.0f;

    for (int nt = 0; nt < 16; ++nt) {
        const int nn = nt * 16 + ncol;
        v8f acc = {};
#pragma unroll
        for (int kk = 0; kk < 32; ++kk) {
            v2f b;
            const float* wp = W1 + (size_t)(kk * 4 + khalf) * DHID + nn;
            b.x = wp[0];
            b.y = wp[DHID];
            acc = __builtin_amdgcn_wmma_f32_16x16x4_f32(
                false, a[kk], false, b, (short)0, acc, false, false);
        }
        const float bias = B1[nn];
        const float w2v  = W2[nn];
#pragma unroll
        for (int v = 0; v < 8; ++v) {
            float h = acc[v] + bias;
            h = h > 0.0f ? h : 0.0f;
            part[v] += h * w2v;            // fused relu(h) @ W2 partial
        }
    }

    // Sum the 16 N-columns held across each 16-lane half (D layout:
    // lanes 0-15 -> rows m0+v, lanes 16-31 -> rows m0+8+v).
#pragma unroll
    for (int v = 0; v < 8; ++v) {
        float s = part[v];
        s += __shfl_xor(s, 8, 32);
        s += __shfl_xor(s, 4, 32);
        s += __shfl_xor(s, 2, 32);
        s += __shfl_xor(s, 1, 32);
        part[v] = s;
    }
    if (lane == 0 || lane == 16) {
        const int base = m0 + (lane >> 4) * 8;
#pragma unroll
        for (int v = 0; v < 8; ++v)
            g_out[base + v] = fabsf(part[v]);
    }
}

// ---------------------------------------------------------------------------
// K2: per-row masked cumsum -> gamma, normalized by row max (0 -> 1).
// One 256-thread block per row, 32 contiguous elements per thread.
// ---------------------------------------------------------------------------
__global__ __launch_bounds__(256) void scan_kernel(
    const float* __restrict__ g,     // (R, L)
    const float* __restrict__ mask,  // (R, L)
    float* __restrict__ gamma)       // (R, L) normalized gamma out
{
    __shared__ float s_sum[256];
    __shared__ float s_red[256];
    __shared__ float s_gmax;

    const int r = blockIdx.x;
    const int t = threadIdx.x;
    const int chunk = LLEN / 256;     // 32
    const int j0 = t * chunk;
    const float* gr = g    + (size_t)r * LLEN;
    const float* mr = mask + (size_t)r * LLEN;
    float* outr     = gamma + (size_t)r * LLEN;

    // Pass 1: per-thread chunk sum of mm*g (col 0 zeroed by col_keep).
    float sum = 0.0f;
    for (int i = 0; i < chunk; ++i) {
        const int j = j0 + i;
        const float mm = (j > 0) ? mr[j] : 0.0f;
        sum += mm * gr[j];
    }
    s_sum[t] = sum;
    __syncthreads();

    // Hillis-Steele inclusive scan over the 256 chunk sums.
    for (int off = 1; off < 256; off <<= 1) {
        const float other = (t >= off) ? s_sum[t - off] : 0.0f;
        __syncthreads();
        s_sum[t] += other;
        __syncthreads();
    }
    float run = (t > 0) ? s_sum[t - 1] : 0.0f;   // exclusive prefix

    // Pass 2: gamma = cumsum(mm*g) * mm, track row max.
    float lmax = -3.4e38f;
    for (int i = 0; i < chunk; ++i) {
        const int j = j0 + i;
        const float mm = (j > 0) ? mr[j] : 0.0f;
        run += mm * gr[j];
        const float gam = run * mm;
        outr[j] = gam;
        lmax = fmaxf(lmax, gam);
    }
    s_red[t] = lmax;
    __syncthreads();
    for (int off = 128; off >= 1; off >>= 1) {
        if (t < off) s_red[t] = fmaxf(s_red[t], s_red[t + off]);
        __syncthreads();
    }
    if (t == 0) {
        const float gm = s_red[0];
        s_gmax = (gm == 0.0f) ? 1.0f : gm;
    }
    __syncthreads();

    // Pass 3: normalize (division, matching reference numerics).
    const float gmax = s_gmax;
    for (int i = 0; i < chunk; ++i) {
        const int j = j0 + i;
        outr[j] = outr[j] / gmax;
    }
}

// ---------------------------------------------------------------------------
// K3: per-(row, segment) reductions: vl, vr -> new_Rl, new_Rr, then A_sum.
// Grid (NSEG, R), 256 threads striding over the row.
// ---------------------------------------------------------------------------
__global__ __launch_bounds__(256) void seg_kernel(
    const float* __restrict__ gamma, // (R, L) normalized
    const float* __restrict__ mask,  // (R, L)
    float* __restrict__ pRl,         // (R*NSEG)
    float* __restrict__ pRr,         // (R*NSEG)
    float* __restrict__ pAs)         // (R*NSEG)
{
    __shared__ float s_a[256];
    __shared__ float s_b[256];
    __shared__ float sh_Rl, sh_Rr;

    const int s = blockIdx.x;
    const int r = blockIdx.y;
    const int t = threadIdx.x;
    const float Rl = (float)s / (float)NSEG;
    const float Rr = (float)(s + 1) / (float)NSEG;
    const float* gr = gamma + (size_t)r * LLEN;
    const float* mr = mask  + (size_t)r * LLEN;

    float vl = -3.4e38f;
    float vr =  3.4e38f;
    for (int j = t; j < LLEN; j += 256) {
        const float gam = gr[j];
        vl = fmaxf(vl, (Rr - gam >= 0.0f) ? gam : 0.0f);
        vr = fminf(vr, (gam - Rl >= 0.0f) ? gam : 10.0f * gam);
    }
    s_a[t] = vl; s_b[t] = vr;
    __syncthreads();
    for (int off = 128; off >= 1; off >>= 1) {
        if (t < off) {
            s_a[t] = fmaxf(s_a[t], s_a[t + off]);
            s_b[t] = fminf(s_b[t], s_b[t + off]);
        }
        __syncthreads();
    }
    if (t == 0) {
        const float nRl = fminf(s_a[0], Rl);
        float tRr = fmaxf(s_b[0], Rr);
        if (s == NSEG - 1) tRr += 0.0001f;
        sh_Rl = nRl; sh_Rr = tRr;
    }
    __syncthreads();
    const float nRl = sh_Rl;
    const float nRr = sh_Rr;

    float asum = 0.0f;
    for (int j = t; j < LLEN; j += 256) {
        const float gam = gr[j];
        const bool inb = (gam - nRl >= 0.0f) && (nRr - gam > 0.0f);
        const float bm = inb ? mr[j] : 0.0f;
        const float A = fmaxf(gam - nRl, 0.0f) + fmaxf(nRr - gam, 0.0f);
        asum += A * bm;
    }
    s_a[t] = asum;
    __syncthreads();
    for (int off = 128; off >= 1; off >>= 1) {
        if (t < off) s_a[t] += s_a[t + off];
        __syncthreads();
    }
    if (t == 0) {
        float As = s_a[0];
        if (As == 0.0f) As = 1.0f;
        const int idx = r * NSEG + s;
        pRl[idx] = nRl; pRr[idx] = nRr; pAs[idx] = As;
    }
}

// ---------------------------------------------------------------------------
// K4: write bound_mask and A_norm (134 MB total) with float4 stores.
// One thread -> 4 consecutive positions of one (row, segment).
// ---------------------------------------------------------------------------
__device__ __forceinline__ void bm_an(float gam, float m, float nRl, float nRr,
                                      float As, float& bm, float& an)
{
    const bool inb = (gam - nRl >= 0.0f) && (nRr - gam > 0.0f);
    bm = inb ? m : 0.0f;
    const float A = fmaxf(gam - nRl, 0.0f) + fmaxf(nRr - gam, 0.0f);
    an = (A * bm) / As;
}

__global__ __launch_bounds__(256) void out_kernel(
    const float* __restrict__ gamma,
    const float* __restrict__ mask,
    const float* __restrict__ pRl,
    const float* __restrict__ pRr,
    const float* __restrict__ pAs,
    float* __restrict__ out_bm,   // (R, NSEG, L)
    float* __restrict__ out_an)   // (R, NSEG, L)
{
    const int tid = blockIdx.x * 256 + threadIdx.x;    // < R*NSEG*L/4 = 4,194,304
    const int qperseg = LLEN / 4;                      // 2048
    const int rs = tid / qperseg;                      // (r*NSEG + s)
    const int j  = (tid % qperseg) * 4;
    const int r  = rs >> 6;

    const float nRl = pRl[rs];
    const float nRr = pRr[rs];
    const float As  = pAs[rs];

    const float4 g4 = *(const float4*)(gamma + (size_t)r * LLEN + j);
    const float4 m4 = *(const float4*)(mask  + (size_t)r * LLEN + j);

    float4 bm, an;
    bm_an(g4.x, m4.x, nRl, nRr, As, bm.x, an.x);
    bm_an(g4.y, m4.y, nRl, nRr, As, bm.y, an.y);
    bm_an(g4.z, m4.z, nRl, nRr, As, bm.z, an.z);
    bm_an(g4.w, m4.w, nRl, nRr, As, bm.w, an.w);

    const size_t o = (size_t)rs * LLEN + j;
    *(float4*)(out_bm + o) = bm;
    *(float4*)(out_an + o) = an;
}

// ---------------------------------------------------------------------------
extern "C" void kernel_launch(void* const* d_in, const int* in_sizes, int n_in,
                              void* d_out, int out_size, void* d_ws, size_t ws_size,
                              hipStream_t stream)
{
    const float* X    = (const float*)d_in[0];  // (8,4,8192,128)
    const float* mask = (const float*)d_in[1];  // (8,4,8192)
    const float* W1   = (const float*)d_in[2];  // (128,256)
    const float* B1   = (const float*)d_in[3];  // (256)
    const float* W2   = (const float*)d_in[4];  // (256,1)

    const int P = RROWS * LLEN;                 // 262144 positions

    // Workspace layout
    float* ws_g     = (float*)d_ws;             // P floats
    float* ws_gamma = ws_g + P;                 // P floats
    float* ws_Rl    = ws_gamma + P;             // R*NSEG
    float* ws_Rr    = ws_Rl + RROWS * NSEG;
    float* ws_As    = ws_Rr + RROWS * NSEG;

    float* out_bm = (float*)d_out;
    float* out_an = out_bm + (size_t)RROWS * NSEG * LLEN;

    // K1: WMMA MLP score. 16 rows per wave, 4 waves per block.
    {
        const int tiles = P / 16;               // 16384
        mlp_score_kernel<<<tiles / 4, 128, 0, stream>>>(X, W1, B1, W2, ws_g, P);
    }
    // K2: per-row scan + normalize.
    scan_kernel<<<RROWS, 256, 0, stream>>>(ws_g, mask, ws_gamma);
    // K3: per-(row, segment) bounds and A_sum.
    {
        dim3 grid(NSEG, RROWS);
        seg_kernel<<<grid, 256, 0, stream>>>(ws_gamma, mask, ws_Rl, ws_Rr, ws_As);
    }
    // K4: write both outputs.
    {
        const int total4 = RROWS * NSEG * LLEN / 4;  // 4,194,304
        out_kernel<<<total4 / 256, 256, 0, stream>>>(ws_gamma, mask, ws_Rl, ws_Rr,
                                                     ws_As, out_bm, out_an);
    }
    (void)in_sizes; (void)n_in; (void)out_size; (void)ws_size;
}